// Meanfield_RNNCell_47768626266242
// MI455X (gfx1250) — compile-verified
//
#include <hip/hip_runtime.h>
#include <math.h>

typedef __attribute__((ext_vector_type(2))) float v2f;
typedef __attribute__((ext_vector_type(8))) float v8f;

#define N_SUPPORTS 4096
#define BATCH      8192
#define ALPHA      0.001f           // DT/TAU = 0.1/100
#define HALF_PI    1.57079632679489662f

// =====================================================================
// Kernel 1: per-support coefficient table (16 floats / support, 256 KB)
// layout per support s (coeff[s*16 + k]):
//   [0..3]  d4, d5, d0, d1                      (delta0 / gains)
//   [4..7]  w*Akm00, w*Akm01, w*Akv00, w*Akv01  (row i=0, B-order [K0,K1,V0,V1])
//   [8..11] w*Akm10, w*Akm11, w*Akv10, w*Akv11  (row i=1)
//   [12..15] w*Wm0, w*Wm1, w*Wv0, w*Wv1         (readout row)
// =====================================================================
__global__ void mf_coeff_kernel(const float* __restrict__ cc,
                                const float* __restrict__ wts,
                                float* __restrict__ coeff) {
    int s = blockIdx.x * blockDim.x + threadIdx.x;
    if (s >= N_SUPPORTS) return;
    float L[7][7];
    const float* p = cc + (size_t)s * 49;
#pragma unroll
    for (int r = 0; r < 7; ++r)
#pragma unroll
        for (int c = 0; c < 7; ++c) L[r][c] = p[r * 7 + c];

    auto dot = [&](int a, int b) {
        float acc = 0.f;
#pragma unroll
        for (int c = 0; c < 7; ++c) acc = fmaf(L[a][c], L[b][c], acc);
        return acc;
    };

    float w = wts[s];
    float* o = coeff + (size_t)s * 16;
    o[0]  = dot(4, 4);      o[1]  = dot(5, 5);
    o[2]  = dot(0, 0);      o[3]  = dot(1, 1);
    o[4]  = w * dot(2, 4);  o[5]  = w * dot(2, 5);
    o[6]  = w * dot(2, 0);  o[7]  = w * dot(2, 1);
    o[8]  = w * dot(3, 4);  o[9]  = w * dot(3, 5);
    o[10] = w * dot(3, 0);  o[11] = w * dot(3, 1);
    o[12] = w * dot(6, 4);  o[13] = w * dot(6, 5);
    o[14] = w * dot(6, 0);  o[15] = w * dot(6, 1);
}

// =====================================================================
// Kernel 2: fused batch x support sweep.
// Block = 256 threads = 8 waves; each wave owns 16 batch rows (128/block).
// Supports DMA'd to LDS (double-buffered, ASYNCcnt) 256 at a time; each
// wave consumes them as 16 tiles of 16 via V_WMMA_F32_16X16X4_F32.
// =====================================================================
#define CHUNK   256
#define LSTRIDE 20      // padded dwords/support in LDS (bank-conflict free)
#define NCHUNKS (N_SUPPORTS / CHUNK)
#define ROWS_PER_BLOCK 128

// 64B global->LDS DMA via CDNA5 async-load path (tracked with ASYNCcnt).
// Per ISA §15.18.3: INST_OFFSET is added to BOTH the global and LDS address,
// so one (lds_addr, global_addr) pair covers all four b128 copies.
__device__ __forceinline__ void async_copy64B(const float* src, float* dst) {
    unsigned lds = (unsigned)(uintptr_t)dst;              // addr[31:0] = LDS byte addr
    unsigned long long ga = (unsigned long long)(uintptr_t)src;
    asm volatile(
        "global_load_async_to_lds_b128 %0, %1, off\n\t"
        "global_load_async_to_lds_b128 %0, %1, off offset:16\n\t"
        "global_load_async_to_lds_b128 %0, %1, off offset:32\n\t"
        "global_load_async_to_lds_b128 %0, %1, off offset:48"
        :: "v"(lds), "v"(ga) : "memory");
}

__device__ __forceinline__ void wait_async0() {
#if __has_builtin(__builtin_amdgcn_s_wait_asynccnt)
    __builtin_amdgcn_s_wait_asynccnt(0);
#else
    asm volatile("s_wait_asynccnt 0x0" ::: "memory");
#endif
}

__global__ __launch_bounds__(256) void mf_main_kernel(
    const float*  __restrict__ input,   // (8192,2)
    const float*  __restrict__ x,       // (8192,4) = [K0,K1,V0,V1]
    const float*  __restrict__ coeff,   // (4096,16)
    float*        __restrict__ out)     // x_next (8192*4) ++ out (8192)
{
    __shared__ float lds_coeff[2 * CHUNK * LSTRIDE];  // 2 x 20 KB (ping-pong)
    __shared__ float lds_red[ROWS_PER_BLOCK * 6];     // 3 KB

    const int tid  = threadIdx.x;
    const int wave = tid >> 5;
    const int lane = tid & 31;
    const int col  = lane & 15;     // N / M index within tile
    const int half = lane >> 4;     // which K-pair this lane carries

    const int row_base = blockIdx.x * ROWS_PER_BLOCK + wave * 16;

    // ---- A fragments (16x4, constant across support loop) ----
    // lanes 0-15: K-indices {0,1}; lanes 16-31: K-indices {2,3}; M = col
    const float4 xr = ((const float4*)x)[row_base + col];
    v2f aX, aD;
    aX.x = half ? xr.z : xr.x;
    aX.y = half ? xr.w : xr.y;
    aD.x = aX.x * aX.x;
    aD.y = aX.y * aX.y;

    v8f accS0 = {}, accS1 = {}, accT0 = {}, accT1 = {}, accU0 = {}, accU1 = {};

    // one 64B support record per thread per chunk
    auto stage = [&](int chunkIdx, int buf) {
        const float* src = coeff + (size_t)(chunkIdx * CHUNK + tid) * 16;
        float*       dst = lds_coeff + buf * (CHUNK * LSTRIDE) + tid * LSTRIDE;
        async_copy64B(src, dst);
    };

    // prologue: DMA chunk 0
    stage(0, 0);
    wait_async0();
    __syncthreads();

    for (int c = 0; c < NCHUNKS; ++c) {
        const int cur = c & 1;
        // kick next chunk's DMA into the other buffer (safe: that buffer was
        // last read in iteration c-1, which ended with a barrier)
        if (c + 1 < NCHUNKS) stage(c + 1, cur ^ 1);

        const float* base = lds_coeff + cur * (CHUNK * LSTRIDE);
        for (int t = 0; t < CHUNK / 16; ++t) {
            const float* cp = base + (t * 16 + col) * LSTRIDE;
            const int h2 = half * 2;
            // B fragments (4x16): VGPR0 = row (0|2), VGPR1 = row (1|3)
            v2f bD, bA0, bA1;
            bD.x  = cp[0 + h2];  bD.y  = cp[1 + h2];
            bA0.x = cp[4 + h2];  bA0.y = cp[5 + h2];
            bA1.x = cp[8 + h2];  bA1.y = cp[9 + h2];
            float wm0 = cp[12], wm1 = cp[13], wv0 = cp[14], wv1 = cp[15];

            v8f cz = {};
            v8f dl = __builtin_amdgcn_wmma_f32_16x16x4_f32(
                false, aD, false, bD, (short)0, cz, false, false);
            v8f n0 = __builtin_amdgcn_wmma_f32_16x16x4_f32(
                false, aX, false, bA0, (short)0, cz, false, false);
            v8f n1 = __builtin_amdgcn_wmma_f32_16x16x4_f32(
                false, aX, false, bA1, (short)0, cz, false, false);

#pragma unroll
            for (int r = 0; r < 8; ++r) {
                float g = __frsqrt_rn(fmaf(HALF_PI, dl[r], 1.0f));
                accS0[r] = fmaf(n0[r], g, accS0[r]);
                accS1[r] = fmaf(n1[r], g, accS1[r]);
                accT0[r] = fmaf(wm0,  g, accT0[r]);
                accT1[r] = fmaf(wm1,  g, accT1[r]);
                accU0[r] = fmaf(wv0,  g, accU0[r]);
                accU1[r] = fmaf(wv1,  g, accU1[r]);
            }
        }

        // make this wave's DMA for chunk c+1 visible, then block-wide sync
        if (c + 1 < NCHUNKS) wait_async0();
        __syncthreads();
    }

    // ---- column (support) reduction: butterfly within each 16-lane half ----
#pragma unroll
    for (int m = 8; m >= 1; m >>= 1) {
#pragma unroll
        for (int r = 0; r < 8; ++r) {
            accS0[r] += __shfl_xor(accS0[r], m, 32);
            accS1[r] += __shfl_xor(accS1[r], m, 32);
            accT0[r] += __shfl_xor(accT0[r], m, 32);
            accT1[r] += __shfl_xor(accT1[r], m, 32);
            accU0[r] += __shfl_xor(accU0[r], m, 32);
            accU1[r] += __shfl_xor(accU1[r], m, 32);
        }
    }
    // C/D layout: VGPR r holds row r (lanes 0-15) / row 8+r (lanes 16-31)
    if (col == 0) {
#pragma unroll
        for (int r = 0; r < 8; ++r) {
            float* q = lds_red + (wave * 16 + half * 8 + r) * 6;
            q[0] = accS0[r]; q[1] = accS1[r];
            q[2] = accT0[r]; q[3] = accT1[r];
            q[4] = accU0[r]; q[5] = accU1[r];
        }
    }
    __syncthreads();

    // ---- final per-row update (one thread per batch row of this block) ----
    if (tid < ROWS_PER_BLOCK) {
        const int b = blockIdx.x * ROWS_PER_BLOCK + tid;
        const float* q = lds_red + tid * 6;
        float S0 = q[0], S1 = q[1], T0 = q[2], T1 = q[3], U0 = q[4], U1 = q[5];

        float4 xv = ((const float4*)x)[b];
        float2 iv = ((const float2*)input)[b];

        float Kn0 = xv.x + ALPHA * (S0 - xv.x);
        float Kn1 = xv.y + ALPHA * (S1 - xv.y);
        float Vn0 = xv.z + ALPHA * (iv.x - xv.z);
        float Vn1 = xv.w + ALPHA * (iv.y - xv.w);

        float4 xn; xn.x = Kn0; xn.y = Kn1; xn.z = Vn0; xn.w = Vn1;
        ((float4*)out)[b] = xn;
        out[(size_t)BATCH * 4 + b] =
            fmaf(Kn0, T0, fmaf(Kn1, T1, fmaf(Vn0, U0, Vn1 * U1)));
    }
}

extern "C" void kernel_launch(void* const* d_in, const int* in_sizes, int n_in,
                              void* d_out, int out_size, void* d_ws, size_t ws_size,
                              hipStream_t stream) {
    const float* input = (const float*)d_in[0];   // (8192,2)
    const float* x     = (const float*)d_in[1];   // (8192,4)
    const float* cc    = (const float*)d_in[2];   // (4096,7,7)
    const float* wts   = (const float*)d_in[3];   // (4096,)
    float*       out   = (float*)d_out;           // 8192*4 ++ 8192
    float*       coeff = (float*)d_ws;            // 4096*16 floats = 256 KB

    mf_coeff_kernel<<<N_SUPPORTS / 256, 256, 0, stream>>>(cc, wts, coeff);
    mf_main_kernel<<<BATCH / ROWS_PER_BLOCK, 256, 0, stream>>>(input, x, coeff, out);
}